// Input_Net_29317446762757
// MI455X (gfx1250) — compile-verified
//
#include <hip/hip_runtime.h>

#define NH        10
#define T_TOTAL   16384
#define S_TOTAL   4096
#define NB        8
#define NC        8
#define WAVES_PB  8
#define CH        64          // sources per async chunk: 64 * 8B = 512B (one b128/lane)
#define DEPTH     8           // async pipeline depth: 4KB in flight per wave
#define CPOL_NT   1           // load temporal hint: NT (read-once 536MB stream > 192MB L2)
#define FLT_BIG   3.402823466e+38f

typedef int v4i __attribute__((vector_size(16)));   // matches builtin param type

#if defined(__gfx1250__) && \
    __has_builtin(__builtin_amdgcn_global_load_async_to_lds_b128) && \
    __has_builtin(__builtin_amdgcn_s_wait_asynccnt)
#define USE_ASYNC 1
#else
#define USE_ASYNC 0
#endif

// Branch-free sorted insert into ascending top-NH lists kept in registers.
#define INSERT10(d2v, sv, bd, bi)                                         \
  do {                                                                    \
    float _d2 = (d2v); int _s = (sv);                                     \
    if (_d2 < bd[NH - 1]) {                                               \
      _Pragma("unroll")                                                   \
      for (int _j = NH - 1; _j >= 1; --_j) {                              \
        bool _shift = _d2 < bd[_j - 1];                                   \
        bool _place = (!_shift) && (_d2 < bd[_j]);                        \
        float _nd = _shift ? bd[_j - 1] : (_place ? _d2 : bd[_j]);        \
        int   _ni = _shift ? bi[_j - 1] : (_place ? _s : bi[_j]);         \
        bd[_j] = _nd; bi[_j] = _ni;                                       \
      }                                                                   \
      bool _p0 = _d2 < bd[0];                                             \
      bi[0] = _p0 ? _s : bi[0];                                           \
      bd[0] = _p0 ? _d2 : bd[0];                                          \
    }                                                                     \
  } while (0)

__global__ __launch_bounds__(WAVES_PB * 32)
void knn_idw_kernel(const float* __restrict__ x,     // [B, S, C]
                    const float* __restrict__ rel,   // [T, S, 2]
                    float* __restrict__ out)         // [B, T, C]
{
#if USE_ASYNC
  __shared__ float4 lbuf[WAVES_PB * DEPTH * (CH / 2)];   // 32 KB / block
#endif
  const int lane = (int)(threadIdx.x & 31u);
  const int wave = (int)(threadIdx.x >> 5u);
  const int t    = (int)blockIdx.x * WAVES_PB + wave;

  float bd[NH]; int bi[NH];
#pragma unroll
  for (int j = 0; j < NH; ++j) { bd[j] = FLT_BIG; bi[j] = 0; }

  const size_t rel_row = (size_t)t * (size_t)S_TOTAL * 2u;   // floats

#if USE_ASYNC
  float4* mybuf = &lbuf[wave * DEPTH * (CH / 2)];
  const int NCHK = S_TOTAL / CH;   // 64 chunks
  // Prologue: fill the async pipeline (each lane moves 16B per chunk).
#pragma unroll
  for (int p = 0; p < DEPTH; ++p) {
    const float* g = rel + rel_row + (size_t)(p * CH + 2 * lane) * 2u;
    __builtin_amdgcn_global_load_async_to_lds_b128(
        (__attribute__((address_space(1))) v4i*)g,
        (__attribute__((address_space(3))) v4i*)&mybuf[p * (CH / 2) + lane],
        0, CPOL_NT);
  }
  for (int ch = 0; ch < NCHK; ++ch) {
    __builtin_amdgcn_s_wait_asynccnt(DEPTH - 1);   // oldest chunk complete
    __asm__ volatile("" ::: "memory");             // order ds reads after wait
    float4 v = mybuf[(ch & (DEPTH - 1)) * (CH / 2) + lane];
    const int s0 = ch * CH + 2 * lane;
    INSERT10(v.x * v.x + v.y * v.y, s0,     bd, bi);
    INSERT10(v.z * v.z + v.w * v.w, s0 + 1, bd, bi);
    const int nch = ch + DEPTH;
    if (nch < NCHK) {
      const float* g = rel + rel_row + (size_t)(nch * CH + 2 * lane) * 2u;
      __builtin_amdgcn_global_load_async_to_lds_b128(
          (__attribute__((address_space(1))) v4i*)g,
          (__attribute__((address_space(3))) v4i*)&mybuf[(nch & (DEPTH - 1)) * (CH / 2) + lane],
          0, CPOL_NT);
    }
  }
  __builtin_amdgcn_s_wait_asynccnt(0);
#else
  // Fallback: direct 16B vector loads + explicit prefetch (global_prefetch_b8).
  const float4* rel4 = (const float4*)(rel + rel_row);
  for (int base = 0; base < S_TOTAL; base += 64) {
    __builtin_prefetch((const void*)(rel4 + (base >> 1) + lane + 512), 0, 1);
    float4 v = rel4[(base >> 1) + lane];
    const int s0 = base + 2 * lane;
    INSERT10(v.x * v.x + v.y * v.y, s0,     bd, bi);
    INSERT10(v.z * v.z + v.w * v.w, s0 + 1, bd, bi);
  }
#endif

  // ---- Wave-wide merge of 32 sorted top-10 lists -> global top-10 ----
  // key = d2_bits(32) | src_idx(12) | lane(5); non-negative floats order as uints;
  // tie-break on source index matches jax.lax.top_k.
  float seld[NH]; int seli[NH];
#pragma unroll
  for (int k = 0; k < NH; ++k) {
    unsigned long long key =
        ((unsigned long long)__float_as_uint(bd[0]) << 32) |
        (unsigned long long)((((unsigned)bi[0] & 0xFFFu) << 5) | (unsigned)lane);
#pragma unroll
    for (int off = 16; off > 0; off >>= 1) {
      unsigned long long o = __shfl_xor(key, off, 32);
      key = (o < key) ? o : key;
    }
    seld[k] = __uint_as_float((unsigned)(key >> 32));
    seli[k] = (int)((key >> 5) & 0xFFFu);
    if (((int)key & 31) == lane) {          // winning lane pops its head
#pragma unroll
      for (int j = 0; j < NH - 1; ++j) { bd[j] = bd[j + 1]; bi[j] = bi[j + 1]; }
      bd[NH - 1] = FLT_BIG; bi[NH - 1] = 0;
    }
  }

  // ---- Normalized inverse-distance weights (redundant per lane, cheap) ----
  float w[NH]; float wsum = 0.0f;
#pragma unroll
  for (int k = 0; k < NH; ++k) {
    w[k] = 1.0f / (sqrtf(seld[k]) + 1e-10f);
    wsum += w[k];
  }
  const float winv = 1.0f / wsum;

  // ---- Gather + weighted sum: lane -> (b0, c) and (b0+4, c) ----
  const int c  = lane & 7;
  const int b0 = lane >> 3;     // 0..3
  float acc0 = 0.0f, acc1 = 0.0f;
#pragma unroll
  for (int k = 0; k < NH; ++k) {
    const float wk = w[k] * winv;
    const int   s  = seli[k];
    acc0 += wk * x[((size_t)b0       * S_TOTAL + (size_t)s) * NC + c];
    acc1 += wk * x[((size_t)(b0 + 4) * S_TOTAL + (size_t)s) * NC + c];
  }
  out[((size_t)b0       * T_TOTAL + (size_t)t) * NC + c] = acc0;
  out[((size_t)(b0 + 4) * T_TOTAL + (size_t)t) * NC + c] = acc1;
}

extern "C" void kernel_launch(void* const* d_in, const int* in_sizes, int n_in,
                              void* d_out, int out_size, void* d_ws, size_t ws_size,
                              hipStream_t stream) {
  const float* x   = (const float*)d_in[0];   // [8, 4096, 8]
  const float* rel = (const float*)d_in[1];   // [16384, 4096, 2]
  float*       out = (float*)d_out;           // [8, 16384, 8]
  (void)in_sizes; (void)n_in; (void)out_size; (void)d_ws; (void)ws_size;

  dim3 grid(T_TOTAL / WAVES_PB);   // 2048
  dim3 block(WAVES_PB * 32);       // 256 (8 wave32 waves)
  hipLaunchKernelGGL(knn_idw_kernel, grid, block, 0, stream, x, rel, out);
}